// RBFN_5901285064816
// MI455X (gfx1250) — compile-verified
//
#include <hip/hip_runtime.h>

typedef __attribute__((ext_vector_type(2))) float v2f;
typedef __attribute__((ext_vector_type(4))) float v4f;
typedef __attribute__((ext_vector_type(8))) float v8f;

#define N_TOTAL   131072
#define C_CENT    512
#define D_IN      64
#define O_OUT     3
#define DC        (D_IN + C_CENT)   // 576
#define LDS_STRIDE 68               // padded center-row stride (floats): lane r -> bank 4r, conflict-free b64 reads
#define ROWS_PER_WAVE 16
#define WAVES_PER_BLOCK 8
#define ROWS_PER_BLOCK (ROWS_PER_WAVE * WAVES_PER_BLOCK)  // 128

__global__ __launch_bounds__(256)
void rbfn_fused_kernel(const float* __restrict__ X,
                       const float* __restrict__ Cent,
                       const float* __restrict__ beta,
                       const float* __restrict__ W,
                       const float* __restrict__ bias,
                       float* __restrict__ out)
{
    extern __shared__ float lds[];
    float* sC    = lds;                              // 512 * 68 floats (padded centers)
    float* sW    = lds + C_CENT * LDS_STRIDE;        // 3 * 576 floats
    float* sBeta = sW + O_OUT * DC;                  // 512 floats

    const int tid = threadIdx.x;

    // ---- Stage centers (512x64, padded to stride 68), W, beta into LDS ----
    for (int i = tid; i < (C_CENT * D_IN) / 4; i += blockDim.x) {
        int r = (i * 4) / D_IN;
        int c = (i * 4) % D_IN;
        v4f v = *(const v4f*)(Cent + r * D_IN + c);
        *(v4f*)(sC + r * LDS_STRIDE + c) = v;
    }
    for (int i = tid; i < O_OUT * DC; i += blockDim.x) sW[i] = W[i];
    for (int i = tid; i < C_CENT;     i += blockDim.x) sBeta[i] = beta[i];
    __syncthreads();

    const int lane = tid & 31;
    const int wave = tid >> 5;
    const int half = lane >> 4;       // 0: K pair {0,1}, 1: K pair {2,3} (f32 A/B WMMA layout)
    const int lrow = lane & 15;       // A: row index; B/C/D: column index
    const int koff = half * 2;
    const int row_base = (blockIdx.x * WAVES_PER_BLOCK + wave) * ROWS_PER_WAVE;

    // ---- Preload the 16 A-fragments for this wave's 16x64 X tile (reused 32x) ----
    const float* xrow = X + (size_t)(row_base + lrow) * D_IN;
    v2f a[16];
    float x2p = 0.0f;
#pragma unroll
    for (int k = 0; k < 16; ++k) {
        a[k] = *(const v2f*)(xrow + k * 4 + koff);
        x2p += a[k].x * a[k].x + a[k].y * a[k].y;
    }
    float x2full = x2p + __shfl_xor(x2p, 16, 32);     // ||x||^2 for row lrow

    // ---- Direct term  X . W[:, 0:64]^T  from the resident A fragments ----
    float dW[O_OUT];
#pragma unroll
    for (int o = 0; o < O_OUT; ++o) {
        float s = 0.0f;
#pragma unroll
        for (int k = 0; k < 16; ++k) {
            s += a[k].x * sW[o * DC + k * 4 + koff]
               + a[k].y * sW[o * DC + k * 4 + koff + 1];
        }
        dW[o] = s + __shfl_xor(s, 16, 32);            // full dot for row lrow
    }

    // x2 broadcast into C/D-layout rows: VGPR i holds row (i + 8*half)
    float x2m[8];
#pragma unroll
    for (int i = 0; i < 8; ++i)
        x2m[i] = __shfl(x2full, i + 8 * half, 32);

    float accW[O_OUT][8];
#pragma unroll
    for (int o = 0; o < O_OUT; ++o)
#pragma unroll
        for (int i = 0; i < 8; ++i) accW[o][i] = 0.0f;

    // ---- Main loop: 32 tiles of 16 centers, K=64 as 16 x (16x16x4 f32 WMMA) ----
    for (int ct = 0; ct < C_CENT / 16; ++ct) {
        const float* crow = sC + (ct * 16 + lrow) * LDS_STRIDE;
        v2f bf[16];
        float c2p = 0.0f;
#pragma unroll
        for (int k = 0; k < 16; ++k) {
            bf[k] = *(const v2f*)(crow + k * 4 + koff);   // ds_load_b64, conflict-free
            c2p += bf[k].x * bf[k].x + bf[k].y * bf[k].y;
        }
        float c2  = c2p + __shfl_xor(c2p, 16, 32);        // ||c||^2 for column lrow
        float bet = sBeta[ct * 16 + lrow];
        float wf0 = sW[0 * DC + D_IN + ct * 16 + lrow];
        float wf1 = sW[1 * DC + D_IN + ct * 16 + lrow];
        float wf2 = sW[2 * DC + D_IN + ct * 16 + lrow];

        v8f acc = {0.f, 0.f, 0.f, 0.f, 0.f, 0.f, 0.f, 0.f};
#pragma unroll
        for (int k = 0; k < 16; ++k) {
            acc = __builtin_amdgcn_wmma_f32_16x16x4_f32(
                      /*neg_a=*/false, a[k], /*neg_b=*/false, bf[k],
                      /*c_mod=*/(short)0, acc, /*reuse_a=*/false, /*reuse_b=*/false);
        }

        // radial = exp(-beta * (||x||^2 + ||c||^2 - 2*cross)), fused into out-proj
#pragma unroll
        for (int i = 0; i < 8; ++i) {
            float sq = x2m[i] + c2 - 2.0f * acc[i];
            float r  = __expf(-bet * sq);
            accW[0][i] += r * wf0;
            accW[1][i] += r * wf1;
            accW[2][i] += r * wf2;
        }
    }

    // ---- Reduce over the 16 column-lanes (stay within each half-wave) ----
#pragma unroll
    for (int o = 0; o < O_OUT; ++o) {
#pragma unroll
        for (int i = 0; i < 8; ++i) {
            float v = accW[o][i];
            v += __shfl_xor(v, 1, 32);
            v += __shfl_xor(v, 2, 32);
            v += __shfl_xor(v, 4, 32);
            v += __shfl_xor(v, 8, 32);
            accW[o][i] = v;
        }
    }

    // Writers: lane j in [0,8) -> row j (half0, VGPR j); lane 24+j -> row 8+j (half1, VGPR j).
    // For both, lane%16 equals the output row, so this lane's dW[] is the right direct term.
    int i = -1;
    if (lane < 8)       i = lane;
    else if (lane >= 24) i = lane - 24;
    if (i >= 0) {
        const int row = row_base + lrow;
        float* po = out + (size_t)row * O_OUT;
        po[0] = accW[0][i] + dW[0] + bias[0];
        po[1] = accW[1][i] + dW[1] + bias[1];
        po[2] = accW[2][i] + dW[2] + bias[2];
    }
}

extern "C" void kernel_launch(void* const* d_in, const int* in_sizes, int n_in,
                              void* d_out, int out_size, void* d_ws, size_t ws_size,
                              hipStream_t stream) {
    const float* X    = (const float*)d_in[0];
    const float* Cent = (const float*)d_in[1];
    const float* beta = (const float*)d_in[2];
    const float* W    = (const float*)d_in[3];
    const float* bias = (const float*)d_in[4];
    float* out = (float*)d_out;

    size_t shmem = (size_t)(C_CENT * LDS_STRIDE + O_OUT * DC + C_CENT) * sizeof(float); // ~148 KB
    dim3 grid(N_TOTAL / ROWS_PER_BLOCK);   // 1024 blocks
    dim3 block(256);                        // 8 wave32s
    rbfn_fused_kernel<<<grid, block, shmem, stream>>>(X, Cent, beta, W, bias, out);
}